// SphericalCKN_44959717655139
// MI455X (gfx1250) — compile-verified
//
#include <hip/hip_runtime.h>
#include <math.h>
#include <stdint.h>

// ---------------------------------------------------------------------------
// SphericalCKN: 3x3 SAME conv (implicit GEMM, fp32 WMMA 16x16x4) + patch-norm
// cosine epilogue.  MI455X / gfx1250: wave32, fp32 WMMA, async global->LDS.
// Each wave: 32 output pixels x 128 filters (2 M-strips share B fragments ->
// 16 WMMAs per 4-wide K-step per 8 LDS fragment loads).
// ---------------------------------------------------------------------------

#define EPS 1e-5f

#define BATCH 32
#define IMGH  64
#define IMGW  64
#define CIN   64
#define FOUT  128
#define TAPS  9                 // 3x3
#define KTOT  576               // 9*64

#define KSTR  132               // padded LDS row stride (floats): 132%64 != 0
                                //  -> K and K+2 rows hit different banks,
                                //  rows stay 16B-aligned for b128 staging
#define CHUNK_FLOATS (CIN * FOUT)   // 8192 floats = 32 KB per tap
#define LDS_BUF_FLOATS (CIN * KSTR) // 8448 floats per buffer

typedef __attribute__((ext_vector_type(2))) float v2f;
typedef __attribute__((ext_vector_type(4))) float v4f;
typedef __attribute__((ext_vector_type(8))) float v8f;
typedef int v4i __attribute__((vector_size(4 * sizeof(int))));  // matches builtin

// ---- CDNA5 async global->LDS path (probe-verified on this toolchain) ----
#if __has_builtin(__builtin_amdgcn_global_load_async_to_lds_b128)
#define USE_ASYNC 1
#else
#define USE_ASYNC 0
#endif

#if USE_ASYNC
#if __has_builtin(__builtin_amdgcn_s_wait_asynccnt)
#define WAIT_ASYNC(n) __builtin_amdgcn_s_wait_asynccnt(n)
#else
#define WAIT_ASYNC(n) asm volatile("s_wait_asynccnt " #n ::: "memory")
#endif
#endif

// Stage one 64x128 fp32 tap of the kernel into a padded LDS buffer.
// 2048 b128 segments, 256 threads -> 8 segments/thread.
__device__ __forceinline__ void stage_chunk(const float* __restrict__ kern,
                                            float* dst, int t, int tid) {
  const float* src = kern + t * CHUNK_FLOATS;
#pragma unroll
  for (int i = 0; i < 8; ++i) {
    int s   = (i << 8) + tid;      // consecutive tids -> consecutive segments
    int row = s >> 5;              // 0..63  (cin row)
    int c4  = (s & 31) << 2;       // 0..124 (filter col, x4 floats)
#if USE_ASYNC
    __builtin_amdgcn_global_load_async_to_lds_b128(
        (__attribute__((address_space(1))) v4i*)(uintptr_t)(src + row * FOUT + c4),
        (__attribute__((address_space(3))) v4i*)(uint32_t)(uintptr_t)(dst + row * KSTR + c4),
        0, 0);
#else
    *(v4f*)(dst + row * KSTR + c4) = *(const v4f*)(src + row * FOUT + c4);
#endif
  }
}

// Prep: inverse kernel norms  inv_kn[f] = 1/sqrt(sum_k k[k][f]^2 + eps)
__global__ void ckn_prep(const float* __restrict__ kern,
                         float* __restrict__ invkn) {
  int f = threadIdx.x;             // 128 threads
  float s = 0.f;
  for (int k = 0; k < KTOT; ++k) { // stride-128 reads: coalesced across f
    float v = kern[k * FOUT + f];
    s += v * v;
  }
  invkn[f] = 1.0f / sqrtf(s + EPS);
}

__global__ __launch_bounds__(256) void ckn_main(const float* __restrict__ x,
                                                const float* __restrict__ kern,
                                                const float* __restrict__ scale,
                                                const float* __restrict__ bias,
                                                const float* __restrict__ invkn,
                                                float* __restrict__ out) {
  extern __shared__ float lds[];   // 2 * LDS_BUF_FLOATS floats (double buffer)

  const int tid   = threadIdx.x;
  const int lane  = tid & 31;
  const int wave  = tid >> 5;      // 0..7
  const int khalf = lane >> 4;     // 0: K{0,1} half, 1: K{2,3} half
  const int fl    = lane & 15;     // filter/row sub-index within tile

  // workgroup = 4 image rows; wave -> (row, 32-pixel half-row)
  const int pr = blockIdx.x * 4 + (wave >> 1);  // 0..2047
  const int b  = pr >> 6;
  const int h  = pr & 63;
  const int w0 = (wave & 1) << 5;               // 0 or 32

  v8f acc0[8], acc1[8];            // strip0: pixels w0+0..15, strip1: +16..31
#pragma unroll
  for (int nt = 0; nt < 8; ++nt) { acc0[nt] = (v8f){}; acc1[nt] = (v8f){}; }
  float esum0 = 0.f, esum1 = 0.f;  // this lane's half of sum(x^2) per row

  stage_chunk(kern, lds, 0, tid);  // prime tap 0 into buffer 0

  for (int t = 0; t < TAPS; ++t) {
    float* cur = lds + (t & 1) * LDS_BUF_FLOATS;
    if (t < TAPS - 1) stage_chunk(kern, lds + ((t + 1) & 1) * LDS_BUF_FLOATS, t + 1, tid);
#if USE_ASYNC
    if (t < TAPS - 1) { WAIT_ASYNC(8); } else { WAIT_ASYNC(0); }
#endif
    __syncthreads();               // tap t resident for all waves

    // --- compute tap t: 16 K-steps of 4 channels, 8 N tiles, 2 M strips ---
    const int dh = t / 3, dw = t % 3;
    const int hy = h + dh - 1;
    const int wx0 = w0 + fl + dw - 1;        // strip0 lane row
    const int wx1 = wx0 + 16;                // strip1 lane row
    const bool hok = (hy >= 0) & (hy < IMGH);
    const bool valid0 = hok & (wx0 >= 0) & (wx0 < IMGW);
    const bool valid1 = hok & (wx1 >= 0) & (wx1 < IMGW);
    const float* ap0 = x + ((((b * IMGH) + hy) * IMGW + wx0) * CIN + (khalf << 1));
    const float* ap1 = ap0 + 16 * CIN;

#pragma unroll 2
    for (int kk = 0; kk < 16; ++kk) {
      float a0x = 0.f, a0y = 0.f, a1x = 0.f, a1y = 0.f;
      if (valid0) { v2f av = *(const v2f*)(ap0 + (kk << 2)); a0x = av.x; a0y = av.y; }
      if (valid1) { v2f av = *(const v2f*)(ap1 + (kk << 2)); a1x = av.x; a1y = av.y; }
      esum0 += a0x * a0x + a0y * a0y;       // patch energy, for free
      esum1 += a1x * a1x + a1y * a1y;
      v2f a0; a0.x = a0x; a0.y = a0y;       // A 16x4: VGPR0=K(2h), VGPR1=K(2h+1)
      v2f a1; a1.x = a1x; a1.y = a1y;

      const float* bp = cur + ((kk << 2) + (khalf << 1)) * KSTR + fl;
#pragma unroll
      for (int nt = 0; nt < 8; ++nt) {
        v2f bf;                              // B 4x16: same K striping as A
        bf.x = bp[nt * 16];
        bf.y = bp[KSTR + nt * 16];
        acc0[nt] = __builtin_amdgcn_wmma_f32_16x16x4_f32(
            false, a0, false, bf, (short)0, acc0[nt], false, false);
        acc1[nt] = __builtin_amdgcn_wmma_f32_16x16x4_f32(
            false, a1, false, bf, (short)0, acc1[nt], false, false);
      }
    }
    __syncthreads();               // all waves done with tap t's buffer
  }

  // --- epilogue: scale * dot / (patch_norm * kernel_norm) + bias ---
  float etot0 = esum0 + __shfl_xor(esum0, 16, 32);  // lane L: energy[row L&15]
  float etot1 = esum1 + __shfl_xor(esum1, 16, 32);
  float ipn0  = 1.0f / sqrtf(etot0 + EPS);
  float ipn1  = 1.0f / sqrtf(etot1 + EPS);
  float sc    = scale[0];

  float ikn[8], bv[8];
#pragma unroll
  for (int nt = 0; nt < 8; ++nt) {
    ikn[nt] = invkn[nt * 16 + fl];
    bv[nt]  = bias[nt * 16 + fl];
  }

  float* obase = out + (((b * IMGH) + h) * IMGW + w0) * FOUT + fl;
#pragma unroll
  for (int r = 0; r < 8; ++r) {
    const int M = r + (khalf << 3);       // C/D layout: VGPR r -> row r or r+8
    float ip0 = __shfl(ipn0, M, 32);      // lane M holds row M's 1/patch_norm
    float ip1 = __shfl(ipn1, M, 32);
    float* orow0 = obase + M * FOUT;
    float* orow1 = obase + (M + 16) * FOUT;
#pragma unroll
    for (int nt = 0; nt < 8; ++nt) {
      orow0[nt * 16] = sc * (acc0[nt][r] * ip0 * ikn[nt]) + bv[nt];
      orow1[nt * 16] = sc * (acc1[nt][r] * ip1 * ikn[nt]) + bv[nt];
    }
  }
}

extern "C" void kernel_launch(void* const* d_in, const int* in_sizes, int n_in,
                              void* d_out, int out_size, void* d_ws, size_t ws_size,
                              hipStream_t stream) {
  (void)in_sizes; (void)n_in; (void)out_size; (void)ws_size;
  const float* x     = (const float*)d_in[0];
  const float* kern  = (const float*)d_in[1];
  const float* scale = (const float*)d_in[2];
  const float* bias  = (const float*)d_in[3];
  float* out   = (float*)d_out;
  float* invkn = (float*)d_ws;            // 128 floats of scratch

  ckn_prep<<<1, 128, 0, stream>>>(kern, invkn);

  const size_t lds_bytes = 2u * LDS_BUF_FLOATS * sizeof(float);  // ~66 KB
  ckn_main<<<BATCH * IMGH / 4, 256, lds_bytes, stream>>>(x, kern, scale, bias,
                                                         invkn, out);
}